// StftToImage_670014898509
// MI455X (gfx1250) — compile-verified
//
#include <hip/hip_runtime.h>

// Shapes from the reference
#define MEL    256
#define CIN    4
#define TFR    512
#define NBINS  1025
#define DOUT   128
#define NBATCH 8
#define MTOTAL (NBATCH * TFR)   // 4096 rows (b,t)
#define MBLK   128              // M rows per workgroup
#define KCH    32               // K chunk staged in LDS
#define LDSTR  34               // padded LDS stride (even -> 8B-aligned b64 frags, bank-conflict-free)
#define KMAX   320              // supports Lmax up to 80 (actual ~33 -> K=132, Kpad=160)

typedef __attribute__((ext_vector_type(2))) float v2f;
typedef __attribute__((ext_vector_type(8))) float v8f;

__global__ __launch_bounds__(256) void mel_group_gemm(
    const float* __restrict__ x,     // [B, C, T, NBINS]
    const float* __restrict__ bank,  // [MEL, Lmax]
    const float* __restrict__ W,     // [MEL, C*Lmax, DOUT]
    const float* __restrict__ bias,  // [MEL, DOUT]
    const int*   __restrict__ idx,   // [MEL, Lmax]
    float*       __restrict__ out,   // [B, DOUT, T, MEL]
    int Lmax)
{
    __shared__ float Alds[MBLK * LDSTR];   // gathered+scaled activations, k-fastest
    __shared__ float Wlds[DOUT * LDSTR];   // W chunk transposed, k-fastest
    __shared__ int   koff[KMAX];           // per-k offset into x: c*T*NBINS + bin
    __shared__ float kscale[KMAX];         // per-k mel weight (0 on padding)

    const int f    = blockIdx.x;           // f fastest in grid -> L2 write-merge over f
    const int mb   = blockIdx.y;
    const int tid  = threadIdx.x;
    const int lane = tid & 31;
    const int wave = tid >> 5;
    const int K    = CIN * Lmax;
    const int Kpad = (K + KCH - 1) & ~(KCH - 1);

    // ---- per-band k-metadata (once per workgroup) ----
    for (int k = tid; k < Kpad; k += 256) {
        int off = 0; float sc = 0.0f;
        if (k < K) {
            int c   = k / Lmax;
            int q   = k - c * Lmax;
            int bin = idx[f * Lmax + q];
            off = c * (TFR * NBINS) + bin;
            sc  = bank[f * Lmax + q];
        }
        koff[k]   = off;
        kscale[k] = sc;
    }
    __syncthreads();

    v8f acc[8] = {};   // 16(M) x 128(N) per wave

    const float* Wf      = W + (size_t)f * (size_t)K * DOUT;
    const int    rowbase = mb * MBLK;

    for (int kb = 0; kb < Kpad; kb += KCH) {
        // ---- stage W chunk transposed: Wlds[n*LDSTR + kk] = W[f][kb+kk][n] ----
        {
            const int kk0 = tid >> 7;      // 0..1
            const int n   = tid & 127;
            #pragma unroll
            for (int i = 0; i < 16; ++i) {
                int kk = kk0 + 2 * i;
                int kg = kb + kk;
                float v = (kg < K) ? Wf[(size_t)kg * DOUT + n] : 0.0f;
                Wlds[n * LDSTR + kk] = v;
            }
        }
        // ---- stage A chunk: Alds[r*LDSTR + kk] = x[b, c, t, bin] * bank ----
        {
            const int kk  = tid & 31;
            const int r0  = tid >> 5;      // 0..7
            const float sc = kscale[kb + kk];
            const int   off = koff[kb + kk];
            #pragma unroll
            for (int i = 0; i < 16; ++i) {
                int r   = r0 + 8 * i;
                int row = rowbase + r;
                int bb  = row >> 9;          // /512
                int tt  = row & (TFR - 1);
                float v = x[bb * (CIN * TFR * NBINS) + tt * NBINS + off] * sc;
                Alds[r * LDSTR + kk] = v;
            }
        }
        __syncthreads();

        // ---- WMMA over this K chunk: 8 k-steps x 8 N-tiles ----
        const int rsel = 16 * wave + (lane & 15);     // A row for this lane
        const int ksel = (lane < 16) ? 0 : 2;         // K-half per lane group
        #pragma unroll
        for (int s = 0; s < 8; ++s) {
            const int K0 = 4 * s;
            // A frag: VGPR0=K0/K0+2, VGPR1=K0+1/K0+3 (even element -> 8B aligned)
            v2f a = *(const v2f*)(&Alds[rsel * LDSTR + K0 + ksel]);
            #pragma unroll
            for (int nt = 0; nt < 8; ++nt) {
                const int n = nt * 16 + (lane & 15);
                v2f bfr = *(const v2f*)(&Wlds[n * LDSTR + K0 + ksel]);
                acc[nt] = __builtin_amdgcn_wmma_f32_16x16x4_f32(
                    /*neg_a=*/false, a, /*neg_b=*/false, bfr,
                    /*c_mod=*/(short)0, acc[nt],
                    /*reuse_a=*/false, /*reuse_b=*/false);
            }
        }
        __syncthreads();
    }

    // ---- epilogue: bias + scatter to out[b,d,t,f] (L2 merges over f) ----
    const int nlo = lane & 15;
    const int mhi = (lane >> 4) * 8;   // VGPR j holds M=j (lanes 0-15) / M=j+8 (lanes 16-31)
    #pragma unroll
    for (int nt = 0; nt < 8; ++nt) {
        const int d  = nt * 16 + nlo;
        const float bv = bias[f * DOUT + d];
        #pragma unroll
        for (int j = 0; j < 8; ++j) {
            int row = rowbase + 16 * wave + mhi + j;
            int bb  = row >> 9;
            int tt  = row & (TFR - 1);
            out[((bb * DOUT + d) * TFR + tt) * MEL + f] = acc[nt][j] + bv;
        }
    }
}

extern "C" void kernel_launch(void* const* d_in, const int* in_sizes, int n_in,
                              void* d_out, int out_size, void* d_ws, size_t ws_size,
                              hipStream_t stream) {
    const float* x    = (const float*)d_in[0];
    const float* bank = (const float*)d_in[1];
    const float* W    = (const float*)d_in[2];
    const float* bvec = (const float*)d_in[3];
    const int*   idx  = (const int*)d_in[4];
    float* out = (float*)d_out;

    const int Lmax = in_sizes[1] / MEL;   // bank is [MEL, Lmax]

    dim3 grid(MEL, MTOTAL / MBLK);        // f fastest -> concurrent f-neighbors share out cachelines
    dim3 block(256);
    mel_group_gemm<<<grid, block, 0, stream>>>(x, bank, W, bvec, idx, out, Lmax);
}